// GCNLayer_31507880083797
// MI455X (gfx1250) — compile-verified
//
#include <hip/hip_runtime.h>

// ---------------------------------------------------------------------------
// GCN layer:  relu( segment_sum( adj_vals * (x @ W)[adj_cols], adj_rows ) )
//   x:       [N=50000, 256] f32
//   W:       [256, 128]     f32
//   edges:   E = 800000 (vals f32, rows/cols i32)
//   out:     [N, 128]       f32
//
// Phase 1 (GEMM) uses V_WMMA_F32_16X16X4_F32 (fp32 precision preserved).
// Phase 2 (SpMM) is the bandwidth-dominant phase; support/output both fit in
// the 192MB L2, so gathers + f32 atomic adds stay on-chip. One wave per edge:
// whole 512B row gathered with a single global_load_b128 per lane-group,
// edge indices scalarized via readfirstlane.
// ---------------------------------------------------------------------------

typedef float v2f __attribute__((ext_vector_type(2)));
typedef float v8f __attribute__((ext_vector_type(8)));

#define IN_F  256
#define OUT_F 128

// ---------------------------------------------------------------------------
// Kernel 1: support = x @ W via fp32 WMMA.
// Block = 256 threads (8 waves). Block -> one 16-row M strip of x.
// Wave w -> N tile [16*w, 16*w+16). 64 k-steps of 16x16x4.
// ---------------------------------------------------------------------------
__global__ __launch_bounds__(256)
void gcn_gemm_wmma_f32(const float* __restrict__ x,
                       const float* __restrict__ w,
                       float* __restrict__ support) {
  const int lane = threadIdx.x & 31;
  const int wave = threadIdx.x >> 5;        // 0..7 -> N tile
  const int m0   = blockIdx.x * 16;         // M tile base (N_NODES % 16 == 0)
  const int n0   = wave * 16;

  const int half = lane >> 4;               // 0: K sub {0,1}; 1: K sub {2,3}
  const int ml   = lane & 15;               // A-row / B-col within tile
  const int kOff = half * 2;

  const float* aRow = x + (size_t)(m0 + ml) * IN_F;
  const float* bCol = w + (size_t)(n0 + ml);

  v8f acc = {};
  #pragma unroll 4
  for (int k = 0; k < IN_F; k += 4) {
    // A fragment (16x4 f32): VGPR0 = A[m][k+kOff], VGPR1 = A[m][k+kOff+1]
    v2f a = *(const v2f*)(aRow + k + kOff);            // 8B contiguous load
    // B fragment (4x16 f32): VGPR0 = B[k+kOff][n], VGPR1 = B[k+kOff+1][n]
    v2f b;
    b.x = bCol[(size_t)(k + kOff)     * OUT_F];
    b.y = bCol[(size_t)(k + kOff + 1) * OUT_F];
    acc = __builtin_amdgcn_wmma_f32_16x16x4_f32(
        /*neg_a=*/false, a, /*neg_b=*/false, b,
        /*c_mod=*/(short)0, acc, /*reuse_a=*/false, /*reuse_b=*/false);
  }

  // C/D layout: VGPR i holds row (i + 8*half), col ml
  float* outBase = support + (size_t)(m0 + half * 8) * OUT_F + n0 + ml;
  #pragma unroll
  for (int i = 0; i < 8; ++i)
    outBase[(size_t)i * OUT_F] = acc[i];
}

// ---------------------------------------------------------------------------
// Kernel 2: edge scatter.  out[rows[e]] += vals[e] * support[cols[e]]
// One wave (32 lanes) per edge; lane covers 4 consecutive columns.
// Gather = one coalesced global_load_b128 per wave (512B = full row).
// Row/col indices scalarized -> saddr-form atomics. 4 f32 atomics per lane.
// support + out stay resident in the 192MB L2.
// ---------------------------------------------------------------------------
__global__ __launch_bounds__(256)
void gcn_edge_scatter(const float* __restrict__ support,
                      const float* __restrict__ vals,
                      const int*   __restrict__ rows,
                      const int*   __restrict__ cols,
                      float* __restrict__ out,
                      int nEdges) {
  const int col4 = (threadIdx.x & 31) << 2;            // 0,4,...,124
  const int sub  = threadIdx.x >> 5;                   // wave in block: 0..7
  const int stride = gridDim.x * 8;
  for (int e = blockIdx.x * 8 + sub; e < nEdges; e += stride) {
    const int   r = __builtin_amdgcn_readfirstlane(rows[e]);
    const int   c = __builtin_amdgcn_readfirstlane(cols[e]);
    const float v = vals[e];
    const float4 g = *(const float4*)(support + (size_t)c * OUT_F + col4);
    float* o = out + (size_t)r * OUT_F + col4;
    atomicAdd(o + 0, v * g.x);
    atomicAdd(o + 1, v * g.y);
    atomicAdd(o + 2, v * g.z);
    atomicAdd(o + 3, v * g.w);
  }
}

// ---------------------------------------------------------------------------
// Kernel 3/0: zero-init and ReLU passes (float4 vectorized).
// ---------------------------------------------------------------------------
__global__ __launch_bounds__(256)
void zero_f4(float4* __restrict__ p, int n4) {
  int i = blockIdx.x * blockDim.x + threadIdx.x;
  if (i < n4) p[i] = make_float4(0.f, 0.f, 0.f, 0.f);
}

__global__ __launch_bounds__(256)
void relu_f4(float4* __restrict__ p, int n4) {
  int i = blockIdx.x * blockDim.x + threadIdx.x;
  if (i < n4) {
    float4 v = p[i];
    v.x = fmaxf(v.x, 0.f);
    v.y = fmaxf(v.y, 0.f);
    v.z = fmaxf(v.z, 0.f);
    v.w = fmaxf(v.w, 0.f);
    p[i] = v;
  }
}

// ---------------------------------------------------------------------------
extern "C" void kernel_launch(void* const* d_in, const int* in_sizes, int n_in,
                              void* d_out, int out_size, void* d_ws, size_t ws_size,
                              hipStream_t stream) {
  const float* x    = (const float*)d_in[0];
  const float* w    = (const float*)d_in[1];
  const float* vals = (const float*)d_in[2];
  const int*   rows = (const int*)  d_in[3];
  const int*   cols = (const int*)  d_in[4];
  float* out     = (float*)d_out;
  float* support = (float*)d_ws;          // N * OUT_F floats = 25.6 MB scratch

  const int nEdges = in_sizes[2];
  const int nNodes = in_sizes[0] / IN_F;  // 50000 (divisible by 16)
  const int total  = nNodes * OUT_F;
  const int n4     = total / 4;

  // 1) zero output accumulator (harness poisons d_out)
  zero_f4<<<(n4 + 255) / 256, 256, 0, stream>>>((float4*)out, n4);

  // 2) support = x @ W   (fp32 WMMA, 8 N-tiles per block)
  gcn_gemm_wmma_f32<<<nNodes / 16, 256, 0, stream>>>(x, w, support);

  // 3) scatter-add over edges (one wave per edge, float4 per lane)
  gcn_edge_scatter<<<12500, 256, 0, stream>>>(support, vals, rows, cols, out, nEdges);

  // 4) relu in place
  relu_f4<<<(n4 + 255) / 256, 256, 0, stream>>>((float4*)out, n4);
}